// PPIModel_6957847020274
// MI455X (gfx1250) — compile-verified
//
#include <hip/hip_runtime.h>
#include <hip/hip_bf16.h>

typedef __attribute__((ext_vector_type(2))) float v2f;
typedef __attribute__((ext_vector_type(8))) float v8f;

#define H 2
#define RNUM 16
#define BNUM 8
#define LNUM 4

// ---------------------------------------------------------------------------
// Basis combine via WMMA: per layer l, W[r][io] = sum_b comp[l][r][b]*V[l][b][io]
// A = comp[l] (16x8 f32), B = V[l] (8x4 f32, zero-padded to 16 cols),
// D = 16x16 f32 (cols 0..3 valid). Two V_WMMA_F32_16X16X4_F32, K = 0..3, 4..7.
// One wave (32 threads) per layer; EXEC all-ones through both WMMAs.
// ---------------------------------------------------------------------------
__global__ void rgcn_basis_wmma(const float* __restrict__ comp,  // [L][16][8]
                                const float* __restrict__ V,     // [L][8][4]
                                float* __restrict__ W)           // [L][16][4]
{
    const int l = blockIdx.x;
    const int t = threadIdx.x;          // 0..31
    const float* C  = comp + (size_t)l * RNUM * BNUM;
    const float* Vb = V    + (size_t)l * BNUM * (H * H);
    const int m  = t & 15;              // A: row M / B,D: col N
    const int hi = (t >> 4) & 1;        // lane half selects K (A/B) or M-half (D)

    v8f acc = {};
    #pragma unroll
    for (int kb = 0; kb < BNUM; kb += 4) {
        const int ka = kb + 2 * hi;     // K index for reg0 in this lane half
        v2f a, b;
        // A 16x4 f32 layout: lanes0-15 reg0/1 = K=kb+0/kb+1; lanes16-31 = kb+2/kb+3
        a[0] = C[m * BNUM + ka];
        a[1] = C[m * BNUM + ka + 1];
        // B 4x16 f32: row K striped across lanes (N = m); only N<4 columns real
        b[0] = (m < (H * H)) ? Vb[ka * (H * H) + m]       : 0.0f;
        b[1] = (m < (H * H)) ? Vb[(ka + 1) * (H * H) + m] : 0.0f;
        acc = __builtin_amdgcn_wmma_f32_16x16x4_f32(
            /*neg_a=*/false, a, /*neg_b=*/false, b,
            /*c_mod=*/(short)0, acc, /*reuse_a=*/false, /*reuse_b=*/false);
    }

    // D 16x16 f32: VGPR j -> row M = j + 8*hi, col N = lane&15
    if (m < (H * H)) {
        float* Wl = W + (size_t)l * RNUM * (H * H);
        #pragma unroll
        for (int j = 0; j < 8; ++j) {
            const int r = j + 8 * hi;
            Wl[r * (H * H) + m] = acc[j];
        }
    }
}

// ---------------------------------------------------------------------------
// Edge kernel: bandwidth-bound gather -> 2x2 matvec -> atomic scatter.
// Indices/norm read as int4/float4 (global_load_b128); relation table in LDS.
// ---------------------------------------------------------------------------
__device__ __forceinline__ void edge_one(int s, int d, int t, float nv,
                                         const float* __restrict__ x,
                                         const float* __restrict__ sW,
                                         float* __restrict__ h)
{
    const float2 xv = ((const float2*)x)[s];
    const float* w = sW + t * 4;        // [W00,W01,W10,W11], io = i*2+o
    const float m0 = (xv.x * w[0] + xv.y * w[2]) * nv;
    const float m1 = (xv.x * w[1] + xv.y * w[3]) * nv;
    atomicAdd(&h[2 * d + 0], m0);
    atomicAdd(&h[2 * d + 1], m1);
}

__global__ void rgcn_edge(const int* __restrict__ es, const int* __restrict__ ed,
                          const int* __restrict__ et, const float* __restrict__ norm,
                          const float* __restrict__ x, const float* __restrict__ Wl,
                          float* __restrict__ h, long long E)
{
    __shared__ float sW[RNUM * H * H];  // 64 floats
    if (threadIdx.x < RNUM * H * H) sW[threadIdx.x] = Wl[threadIdx.x];
    __syncthreads();

    const long long n4     = (E + 3) >> 2;
    const long long stride = (long long)gridDim.x * blockDim.x;
    for (long long i = (long long)blockIdx.x * blockDim.x + threadIdx.x;
         i < n4; i += stride) {
        const long long e0 = i << 2;
        if (e0 + 3 < E) {
            const int4   s4 = ((const int4*)es)[i];
            const int4   d4 = ((const int4*)ed)[i];
            const int4   t4 = ((const int4*)et)[i];
            const float4 nv = ((const float4*)norm)[i];
            edge_one(s4.x, d4.x, t4.x, nv.x, x, sW, h);
            edge_one(s4.y, d4.y, t4.y, nv.y, x, sW, h);
            edge_one(s4.z, d4.z, t4.z, nv.z, x, sW, h);
            edge_one(s4.w, d4.w, t4.w, nv.w, x, sW, h);
        } else {
            for (long long e = e0; e < E; ++e)
                edge_one(es[e], ed[e], et[e], norm[e], x, sW, h);
        }
    }
}

// ---------------------------------------------------------------------------
// Node kernel: h + bias, optional relu, optional skip add; float2 traffic.
// ---------------------------------------------------------------------------
__global__ void rgcn_node(const float* __restrict__ h, const float* __restrict__ bias,
                          int layer, int do_relu, int add_skip,
                          const float* __restrict__ skip,
                          float* __restrict__ xout, int N)
{
    const float b0 = bias[layer * H + 0];
    const float b1 = bias[layer * H + 1];
    const int stride = gridDim.x * blockDim.x;
    for (int i = blockIdx.x * blockDim.x + threadIdx.x; i < N; i += stride) {
        float2 v = ((const float2*)h)[i];
        v.x += b0; v.y += b1;
        if (do_relu) { v.x = fmaxf(v.x, 0.0f); v.y = fmaxf(v.y, 0.0f); }
        if (add_skip) {
            const float2 s = ((const float2*)skip)[i];
            v.x += s.x; v.y += s.y;
        }
        ((float2*)xout)[i] = v;
    }
}

__global__ void zero_f32(float* __restrict__ p, long long n)
{
    const long long stride = (long long)gridDim.x * blockDim.x;
    for (long long i = (long long)blockIdx.x * blockDim.x + threadIdx.x;
         i < n; i += stride) p[i] = 0.0f;
}

// ---------------------------------------------------------------------------
// Final MLP: dot(x, w_mlp) over 2N elems -> sigmoid. Wave32 shuffle reduce.
// ---------------------------------------------------------------------------
__global__ void dot_reduce(const float* __restrict__ x, const float* __restrict__ w,
                           long long n, float* __restrict__ accum)
{
    float s = 0.0f;
    const long long stride = (long long)gridDim.x * blockDim.x;
    for (long long i = (long long)blockIdx.x * blockDim.x + threadIdx.x;
         i < n; i += stride) s += x[i] * w[i];
    #pragma unroll
    for (int off = 16; off > 0; off >>= 1) s += __shfl_down(s, off, 32);
    if ((threadIdx.x & 31) == 0) atomicAdd(accum, s);
}

__global__ void sigmoid_out(const float* __restrict__ accum,
                            const float* __restrict__ b_mlp,
                            float* __restrict__ out)
{
    const float z = accum[0] + b_mlp[0];
    out[0] = 1.0f / (1.0f + expf(-z));
}

// ---------------------------------------------------------------------------
extern "C" void kernel_launch(void* const* d_in, const int* in_sizes, int n_in,
                              void* d_out, int out_size, void* d_ws, size_t ws_size,
                              hipStream_t stream)
{
    const float* features = (const float*)d_in[0];   // [2N]
    const float* norm     = (const float*)d_in[1];   // [E]
    const float* V        = (const float*)d_in[2];   // [L][8][2][2]
    const float* comp     = (const float*)d_in[3];   // [L][16][8]
    const float* bias     = (const float*)d_in[4];   // [L][2]
    const float* w_mlp    = (const float*)d_in[5];   // [2N]
    const float* b_mlp    = (const float*)d_in[6];   // [1]
    const int*   es       = (const int*)d_in[7];     // [E]
    const int*   ed       = (const int*)d_in[8];     // [E]
    const int*   et       = (const int*)d_in[9];     // [E]

    const long long E = (long long)in_sizes[7];
    const int N = in_sizes[0] / H;
    const long long twoN = 2LL * N;

    float* ws    = (float*)d_ws;
    float* Wtab  = ws;                 // L*16*4 = 256 floats
    float* accum = ws + 256;           // 1 float (padded)
    float* bufH  = ws + 512;           // 2N floats (scatter target)
    float* buf1  = bufH + twoN;        // 2N floats
    float* buf2  = buf1 + twoN;        // 2N floats

    const int TB = 256;
    const int edgeBlocks = 8192;
    const int nodeBlocks = (N + TB - 1) / TB;
    const int zeroBlocks = 2048;

    // 1) all relation weight tables via WMMA (one wave per layer)
    rgcn_basis_wmma<<<LNUM, 32, 0, stream>>>(comp, V, Wtab);

    // layer 0: x1 = relu(h0 + b0)
    zero_f32<<<zeroBlocks, TB, 0, stream>>>(bufH, twoN);
    rgcn_edge<<<edgeBlocks, TB, 0, stream>>>(es, ed, et, norm, features,
                                             Wtab + 0 * 64, bufH, E);
    rgcn_node<<<nodeBlocks, TB, 0, stream>>>(bufH, bias, 0, 1, 0, nullptr, buf1, N);

    // layer 1: x2 = relu(h1 + b1) + features ; skip = x2
    zero_f32<<<zeroBlocks, TB, 0, stream>>>(bufH, twoN);
    rgcn_edge<<<edgeBlocks, TB, 0, stream>>>(es, ed, et, norm, buf1,
                                             Wtab + 1 * 64, bufH, E);
    rgcn_node<<<nodeBlocks, TB, 0, stream>>>(bufH, bias, 1, 1, 1, features, buf2, N);

    // layer 2: x3 = relu(h2 + b2)
    zero_f32<<<zeroBlocks, TB, 0, stream>>>(bufH, twoN);
    rgcn_edge<<<edgeBlocks, TB, 0, stream>>>(es, ed, et, norm, buf2,
                                             Wtab + 2 * 64, bufH, E);
    rgcn_node<<<nodeBlocks, TB, 0, stream>>>(bufH, bias, 2, 1, 0, nullptr, buf1, N);

    // layer 3: x4 = (h3 + b3) + x2   (no relu; skip == buf2)
    zero_f32<<<zeroBlocks, TB, 0, stream>>>(bufH, twoN);
    rgcn_edge<<<edgeBlocks, TB, 0, stream>>>(es, ed, et, norm, buf1,
                                             Wtab + 3 * 64, bufH, E);
    rgcn_node<<<nodeBlocks, TB, 0, stream>>>(bufH, bias, 3, 0, 1, buf2, buf1, N);

    // final MLP: sigmoid(dot(x4, w_mlp) + b_mlp)
    zero_f32<<<1, 32, 0, stream>>>(accum, 1);
    dot_reduce<<<1024, TB, 0, stream>>>(buf1, w_mlp, twoN, accum);
    sigmoid_out<<<1, 1, 0, stream>>>(accum, b_mlp, (float*)d_out);
}